// RCNNSparse3DLoss_15719580304287
// MI455X (gfx1250) — compile-verified
//
#include <hip/hip_runtime.h>
#include <math.h>

// ---------------------------------------------------------------------------
// RCNNSparse3DLoss for MI455X (gfx1250).
// B=8, P=512, W=H=32, F=1024. Memory-bound argmax+gather (16 MB stream) with
// a per-pair scalar geometry tail. One wave32 per (b,p) pair; cls scores are
// staged global->LDS with the CDNA5 async copy engine (ASYNCcnt) and reduced
// with wave32 shuffles.
// ---------------------------------------------------------------------------

#define B_DIM 8
#define P_DIM 512
#define W_DIM 32
#define F_DIM 1024                    // 32*32
#define PAIRS_PER_BLOCK 8             // one wave32 per pair
#define BLOCK_THREADS 256             // 8 waves
#define NUM_PAIRS (B_DIM * P_DIM)     // 4096
#define GRID_BLOCKS (NUM_PAIRS / PAIRS_PER_BLOCK)  // 512
#define PI_F 3.14159265358979323846f

__global__ __launch_bounds__(BLOCK_THREADS)
void rcnn_sparse3d_loss_kernel(
    const float* __restrict__ batch_loc,    // (B,P,3)
    const float* __restrict__ batch_dim,    // (B,P,3)  h,w,l
    const float* __restrict__ batch_rot_y,  // (B,P,1)
    const float* __restrict__ rois,         // (B,P,4)
    const float* __restrict__ cls,          // (B,P,1,32,32)
    const float* __restrict__ off2d,        // (B,P,2,32,32)
    const float* __restrict__ off3d,        // (B,P,2,32,32)
    const float* __restrict__ depth_map,    // (B,P,1,32,32)
    const float* __restrict__ pre_dim,      // (B,P,3)  h,w,l
    const float* __restrict__ pre_rot,      // (B,P,2)  sin,cos
    const float* __restrict__ calib,        // (B,3,4)
    const float* __restrict__ trans_mat,    // (B,2,3)
    const float* __restrict__ dist,         // (B,5)
    float* __restrict__ out)                // (B,P,1)
{
    __shared__ float smem[PAIRS_PER_BLOCK * F_DIM];   // 32 KB / block

    const int tid   = threadIdx.x;
    const int lane  = tid & 31;
    const int wave  = tid >> 5;
    const int pair0 = blockIdx.x * PAIRS_PER_BLOCK;   // first (b,p) of block

    // ---- Stage the block's 8 contiguous cls tiles (32 KB) into LDS --------
#if defined(__gfx1250__)
    {
        // GVS mode: mem = SADDR(64b) + VADDR(32b offset); dest = LDS byte addr.
        const unsigned long long gbase =
            (unsigned long long)(const void*)(cls + (size_t)pair0 * F_DIM);
        const unsigned lds_base = (unsigned)(unsigned long long)(&smem[0]);
        #pragma unroll
        for (int j = 0; j < 8; ++j) {
            const unsigned boff    = (unsigned)((j * BLOCK_THREADS + tid) * 16);
            const unsigned ldsaddr = lds_base + boff;
            asm volatile("global_load_async_to_lds_b128 %0, %1, %2"
                         :: "v"(ldsaddr), "v"(boff), "s"(gbase)
                         : "memory");
        }
        asm volatile("s_wait_asynccnt 0" ::: "memory");   // drain ASYNCcnt
    }
#else
    {
        const float4* gsrc = (const float4*)(cls + (size_t)pair0 * F_DIM);
        float4*       ldst = (float4*)smem;
        for (int j = tid; j < PAIRS_PER_BLOCK * F_DIM / 4; j += BLOCK_THREADS)
            ldst[j] = gsrc[j];
    }
#endif
    __syncthreads();

    // ---- Wave32 argmax over this wave's 1024 scores (first-max semantics) --
    const float4* region = (const float4*)(smem + wave * F_DIM);
    float bestv = -INFINITY;
    int   besti = 0;
    #pragma unroll
    for (int k = 0; k < 8; ++k) {
        const float4 v  = region[k * 32 + lane];   // ds_load_b128
        const int    fb = (k * 32 + lane) * 4;
        if (v.x > bestv) { bestv = v.x; besti = fb + 0; }
        if (v.y > bestv) { bestv = v.y; besti = fb + 1; }
        if (v.z > bestv) { bestv = v.z; besti = fb + 2; }
        if (v.w > bestv) { bestv = v.w; besti = fb + 3; }
    }
    #pragma unroll
    for (int m = 16; m >= 1; m >>= 1) {
        const float ov = __shfl_xor(bestv, m, 32);
        const int   oi = __shfl_xor(besti, m, 32);
        if (ov > bestv || (ov == bestv && oi < besti)) { bestv = ov; besti = oi; }
    }

    // ---- Per-pair scalar tail (lane 0 of each wave) ------------------------
    if (lane == 0) {
        const int gp = pair0 + wave;          // 0..4095
        const int b  = gp >> 9;               // / P_DIM
        const int f  = besti;

        const size_t base1 = (size_t)gp * F_DIM;
        const size_t base2 = (size_t)gp * 2 * F_DIM;
        const float g2x  = off2d[base2 + f];
        const float g2y  = off2d[base2 + F_DIM + f];
        const float g3x  = off3d[base2 + f];
        const float g3y  = off3d[base2 + F_DIM + f];
        const float gdep = depth_map[base1 + f];

        const float ixf = (float)(f & 31);    // idx % W
        const float iyf = (float)(f >> 5);    // floor(idx / W)

        const float* R = rois + (size_t)gp * 4;
        const float sclx = (float)W_DIM / (R[2] - R[0] + 1.0f);
        const float scly = (float)W_DIM / (R[3] - R[1] + 1.0f);
        const float cx2  = (ixf + g2x) / sclx + R[0];
        const float cy2  = (iyf + g2y) / scly + R[1];

        // inverse of [[a,b,c],[d,e,f],[0,0,1]]
        const float* TM = trans_mat + (size_t)b * 6;
        const float ta = TM[0], tb = TM[1], tc = TM[2];
        const float td = TM[3], te = TM[4], tf = TM[5];
        const float inv = 1.0f / (ta * te - tb * td);
        const float ax = ( te * cx2 - tb * cy2 + (tb * tf - tc * te)) * inv;
        const float ay = (-td * cx2 + ta * cy2 + (tc * td - ta * tf)) * inv;

        // undistort (full iteration; harness uses zero dist but keep general)
        const float* C = calib + (size_t)b * 12;
        const float fx = C[0], cxx = C[2], fy = C[5], cyy = C[6];
        const float* D = dist + (size_t)b * 5;
        const float k1 = D[0], k2 = D[1], p1 = D[2], p2 = D[3], k3 = D[4];
        const float x0 = (ax - cxx) / fx;
        const float y0 = (ay - cyy) / fy;
        float x = x0, y = y0;
        #pragma unroll
        for (int it = 0; it < 5; ++it) {
            const float r2  = x * x + y * y;
            const float icd = 1.0f / (1.0f + ((k3 * r2 + k2) * r2 + k1) * r2);
            const float dx  = 2.0f * p1 * x * y + p2 * (r2 + 2.0f * x * x);
            const float dy  = p1 * (r2 + 2.0f * y * y) + 2.0f * p2 * x * y;
            x = (x0 - dx) * icd;
            y = (y0 - dy) * icd;
        }
        const float ux = fx * x + cxx;
        const float uy = fy * y + cyy;

        const float fscale = fx * (1.0f / 1000.0f);
        const float sig    = 1.0f / (1.0f + expf(-gdep));
        const float depth  = (1.0f / (sig + 1e-6f) - 1.0f) * fscale;
        const float z3     = depth - C[11];
        float x3 = (ux * depth - C[3] - C[2] * z3) / C[0];
        float y3 = (uy * depth - C[7] - C[6] * z3) / C[5];
        x3 += g3x * fscale;
        y3 += g3y * fscale;

        const float* PD = pre_dim + (size_t)gp * 3;
        const float locx = x3;
        const float locz = z3;
        (void)y3;  // y enters only the unused (BEV drops y) corner coordinate

        const float* PR = pre_rot + (size_t)gp * 2;
        const float rs = PR[0], rc = PR[1];
        float alpha = atanf(rs / (rc + 1e-7f))
                    + (rc < 0.0f ? 1.0f : -1.0f) * (PI_F * 0.5f);
        const float theta = atanf(locx / (locz + 1e-7f));
        float ry = alpha + theta;
        ry += ((ry < -PI_F ? 1.0f : 0.0f) - (ry > PI_F ? 1.0f : 0.0f))
              * (2.0f * PI_F);

        // pred BEV AABB: corners (+-l/2, +-w/2) rotated -> half-extents
        const float pw = PD[1], pl = PD[2];
        const float pc = cosf(ry), ps = sinf(ry);
        const float phx = fabsf(pc) * (pl * 0.5f) + fabsf(ps) * (pw * 0.5f);
        const float phz = fabsf(ps) * (pl * 0.5f) + fabsf(pc) * (pw * 0.5f);
        const float pminx = locx - phx, pmaxx = locx + phx;
        const float pminz = locz - phz, pmaxz = locz + phz;

        // gt BEV AABB
        const float* GL = batch_loc + (size_t)gp * 3;
        const float* GD = batch_dim + (size_t)gp * 3;
        const float gry = batch_rot_y[gp];
        const float gc = cosf(gry), gs = sinf(gry);
        const float ghx = fabsf(gc) * (GD[2] * 0.5f) + fabsf(gs) * (GD[1] * 0.5f);
        const float ghz = fabsf(gs) * (GD[2] * 0.5f) + fabsf(gc) * (GD[1] * 0.5f);
        const float gminx = GL[0] - ghx, gmaxx = GL[0] + ghx;
        const float gminz = GL[2] - ghz, gmaxz = GL[2] + ghz;

        const float s1 = (pmaxx - pminx) * (pmaxz - pminz);
        const float s2 = (gmaxx - gminx) * (gmaxz - gminz);
        const float wov = fmaxf(fminf(pmaxx, gmaxx) - fmaxf(pminx, gminx), 0.0f);
        const float hov = fmaxf(fminf(pmaxz, gmaxz) - fmaxf(pminz, gminz), 0.0f);
        const float ovl = wov * hov;
        out[gp] = ovl / (s1 + s2 - ovl + 1e-6f);
    }
}

extern "C" void kernel_launch(void* const* d_in, const int* in_sizes, int n_in,
                              void* d_out, int out_size, void* d_ws, size_t ws_size,
                              hipStream_t stream) {
    (void)in_sizes; (void)n_in; (void)out_size; (void)d_ws; (void)ws_size;
    rcnn_sparse3d_loss_kernel<<<GRID_BLOCKS, BLOCK_THREADS, 0, stream>>>(
        (const float*)d_in[0],   // batch_loc
        (const float*)d_in[1],   // batch_dim
        (const float*)d_in[2],   // batch_rot_y
        (const float*)d_in[3],   // rois
        (const float*)d_in[4],   // pre_cls_map
        (const float*)d_in[5],   // pre_2d_offset
        (const float*)d_in[6],   // pre_3d_offset
        (const float*)d_in[7],   // pre_depth
        (const float*)d_in[8],   // pre_dim
        (const float*)d_in[9],   // pre_rot
        (const float*)d_in[10],  // calib
        (const float*)d_in[11],  // trans_mat
        (const float*)d_in[12],  // dist_coeffs
        (float*)d_out);
}